// ScaledDotProductAttention_47768626266156
// MI455X (gfx1250) — compile-verified
//
#include <hip/hip_runtime.h>
#include <math.h>

// CDNA5 / gfx1250: wave32, WMMA f32 16x16x4 path.
typedef __attribute__((ext_vector_type(2))) float v2f;
typedef __attribute__((ext_vector_type(8))) float v8f;

#define WMMA_F32(A, B, C) \
  __builtin_amdgcn_wmma_f32_16x16x4_f32(false, (A), false, (B), (short)0, (C), false, false)

constexpr int BB = 4;
constexpr int HH = 16;
constexpr int SS = 2048;
constexpr int DD = 64;
constexpr int KT = SS / 16;           // 128 k-tiles
constexpr float INV_T = 0.125f;       // 1 / TEMPERATURE
constexpr float NEG_BIG = -1e30f;
constexpr int PST = 20;               // padded LDS row stride (floats)

__global__ __launch_bounds__(32)
void ScaledDotProductAttention_47768626266156_kernel(
    const float* __restrict__ q, const float* __restrict__ k,
    const float* __restrict__ v, const int* __restrict__ mask,
    float* __restrict__ out, float* __restrict__ attn)
{
  __shared__ float pbuf[16 * PST];

  const int lane = threadIdx.x;         // 0..31
  const int lo   = lane & 15;
  const int half = lane >> 4;           // 0 or 1
  const int tile = blockIdx.x;          // b*H*128 + h*128 + qt
  const int qt = tile & (KT - 1);
  const int h  = (tile >> 7) & (HH - 1);
  const int b  = tile >> 11;

  const long bh    = (long)(b * HH + h);
  const float* qrow = q + (bh * SS + qt * 16 + lo) * DD;   // this lane's Q row
  const float* kbase = k + bh * SS * DD;
  const float* vbase = v + bh * SS * DD;
  const int*   mbase = mask + ((long)b * SS + qt * 16) * SS;

  // ---- Preload Q A-fragments (ISA 16x4 f32 A layout), fold in 1/temperature.
  // Chunk c covers K-dim d = 4c..4c+3; this lane holds d = 4c + 2*half + {0,1}.
  float aq[32];
#pragma unroll
  for (int c = 0; c < 16; ++c) {
    v2f t = *(const v2f*)(qrow + 4 * c + 2 * half);
    aq[2 * c]     = t.x * INV_T;
    aq[2 * c + 1] = t.y * INV_T;
  }

  // Online-softmax state. C/D layout: vgpr r <-> row (r + 8*half), col = lo.
  float mrun[8], lrun[8];
  v8f acc[4];
#pragma unroll
  for (int r = 0; r < 8; ++r) { mrun[r] = NEG_BIG; lrun[r] = 0.0f; }
#pragma unroll
  for (int nt = 0; nt < 4; ++nt) acc[nt] = (v8f){};

  // ================= Pass 1: flash-style O accumulation =================
  for (int kt = 0; kt < KT; ++kt) {
    const float* krow = kbase + (kt * 16 + lo) * DD;
    if (kt + 1 < KT) __builtin_prefetch(krow + 16 * DD, 0, 1);  // global_prefetch_b8

    // S = (Q/T) * K^T : 16 chained v_wmma_f32_16x16x4_f32 over D=64.
    // B fragment (K^T) has the same per-lane pattern as A: row lo, d=4c+2*half.
    v8f s = (v8f){};
#pragma unroll
    for (int c = 0; c < 16; ++c) {
      v2f a; a.x = aq[2 * c]; a.y = aq[2 * c + 1];
      v2f bb = *(const v2f*)(krow + 4 * c + 2 * half);
      s = WMMA_F32(a, bb, s);
    }

    // Mask: mask[b][qrow][kcol] == 0 -> -inf
    const int* mcol = mbase + kt * 16 + lo;
#pragma unroll
    for (int r = 0; r < 8; ++r) {
      int mv = mcol[(long)(r + 8 * half) * SS];
      s[r] = mv ? s[r] : -INFINITY;
    }

    // Row max / rescale / exp / row sum (rows live across the 16-lane half).
    float fac[8];
#pragma unroll
    for (int r = 0; r < 8; ++r) {
      float mx = s[r];
      mx = fmaxf(mx, __shfl_xor(mx, 1, 32));
      mx = fmaxf(mx, __shfl_xor(mx, 2, 32));
      mx = fmaxf(mx, __shfl_xor(mx, 4, 32));
      mx = fmaxf(mx, __shfl_xor(mx, 8, 32));
      float mn = fmaxf(fmaxf(mrun[r], mx), NEG_BIG);
      fac[r] = __expf(mrun[r] - mn);
      mrun[r] = mn;
      float p = __expf(s[r] - mn);      // masked: exp(-inf) = 0
      s[r] = p;
      float ps = p;
      ps += __shfl_xor(ps, 1, 32);
      ps += __shfl_xor(ps, 2, 32);
      ps += __shfl_xor(ps, 4, 32);
      ps += __shfl_xor(ps, 8, 32);
      lrun[r] = lrun[r] * fac[r] + ps;
    }
#pragma unroll
    for (int nt = 0; nt < 4; ++nt)
#pragma unroll
      for (int r = 0; r < 8; ++r) acc[nt][r] *= fac[r];

    // Re-fragment P (D-layout -> A-layout) via LDS.
    __syncthreads();
#pragma unroll
    for (int r = 0; r < 8; ++r) pbuf[(r + 8 * half) * PST + lo] = s[r];
    __syncthreads();

    v2f pa[4];
#pragma unroll
    for (int c = 0; c < 4; ++c)
      pa[c] = *(const v2f*)(&pbuf[lo * PST + 4 * c + 2 * half]);

    // O += P * V : 4 column tiles x 4 chained WMMAs (K-dim = 16).
#pragma unroll
    for (int nt = 0; nt < 4; ++nt) {
      v8f a = acc[nt];
#pragma unroll
      for (int c = 0; c < 4; ++c) {
        const float* vcol = vbase + (long)(kt * 16 + 4 * c + 2 * half) * DD + nt * 16 + lo;
        v2f bb; bb.x = vcol[0]; bb.y = vcol[DD];
        a = WMMA_F32(pa[c], bb, a);
      }
      acc[nt] = a;
    }
  }

  // Normalize and write O. Write-once stream -> non-temporal stores
  // (keep L2 for mask/K/V which are reused across heads / q-tiles).
  float rcp[8];
#pragma unroll
  for (int r = 0; r < 8; ++r) rcp[r] = 1.0f / fmaxf(lrun[r], 1e-37f);

  float* orow = out + (bh * SS + qt * 16) * DD;
#pragma unroll
  for (int nt = 0; nt < 4; ++nt)
#pragma unroll
    for (int r = 0; r < 8; ++r)
      __builtin_nontemporal_store(acc[nt][r] * rcp[r],
                                  orow + (long)(r + 8 * half) * DD + nt * 16 + lo);

  // ================= Pass 2: recompute scores, emit normalized attn =====
  float* arow = attn + (bh * SS + qt * 16) * (long)SS;
  for (int kt = 0; kt < KT; ++kt) {
    const float* krow = kbase + (kt * 16 + lo) * DD;
    if (kt + 1 < KT) __builtin_prefetch(krow + 16 * DD, 0, 1);

    v8f s = (v8f){};
#pragma unroll
    for (int c = 0; c < 16; ++c) {
      v2f a; a.x = aq[2 * c]; a.y = aq[2 * c + 1];
      v2f bb = *(const v2f*)(krow + 4 * c + 2 * half);
      s = WMMA_F32(a, bb, s);
    }

    const int* mcol = mbase + kt * 16 + lo;
#pragma unroll
    for (int r = 0; r < 8; ++r) {
      int mv = mcol[(long)(r + 8 * half) * SS];
      float sv = mv ? s[r] : -INFINITY;
      float p = __expf(sv - mrun[r]) * rcp[r];   // masked -> 0
      // 1.07 GB write-once stream: non-temporal, coalesced 64B per vgpr row.
      __builtin_nontemporal_store(p, arow + (long)(r + 8 * half) * SS + kt * 16 + lo);
    }
  }
}

extern "C" void kernel_launch(void* const* d_in, const int* in_sizes, int n_in,
                              void* d_out, int out_size, void* d_ws, size_t ws_size,
                              hipStream_t stream) {
  const float* q   = (const float*)d_in[0];
  const float* k   = (const float*)d_in[1];
  const float* v   = (const float*)d_in[2];
  const int*  mask = (const int*)d_in[3];

  float* out  = (float*)d_out;                               // [B,H,S,D]
  float* attn = out + (size_t)BB * HH * SS * DD;             // [B,H,S,S]

  dim3 grid(BB * HH * KT);   // 8192 workgroups, one wave32 each
  ScaledDotProductAttention_47768626266156_kernel<<<grid, 32, 0, stream>>>(
      q, k, v, mask, out, attn);
}